// SudokuPolicy_2190433321671
// MI455X (gfx1250) — compile-verified
//
#include <hip/hip_runtime.h>

#define N_NODES   50000
#define N_EDGES   800000
#define HID       96
#define N_LAYERS  4
#define N_ACTIONS 729
#define WT_STRIDE 100   // padded LDS stride (dwords): conflict-free B-fragment loads

typedef __attribute__((ext_vector_type(2))) float v2f;
typedef __attribute__((ext_vector_type(8))) float v8f;

static __device__ __forceinline__ void atomic_add_f32(float* p, float v) {
  // relaxed, device scope -> lowers to global_atomic_add_f32 (no return)
  __hip_atomic_fetch_add(p, v, __ATOMIC_RELAXED, __HIP_MEMORY_SCOPE_AGENT);
}

// ---------------- degree / normalization (computed once; layer-invariant) --------------
__global__ void k_deg_init(float* __restrict__ deg) {
  int i = blockIdx.x * blockDim.x + threadIdx.x;
  if (i < N_NODES) deg[i] = 1.0f;  // self-loop contributes 1
}

__global__ void k_deg_accum(const long long* __restrict__ dst, float* __restrict__ deg) {
  int e = blockIdx.x * blockDim.x + threadIdx.x;
  if (e < N_EDGES) atomic_add_f32(&deg[(int)dst[e]], 1.0f);
}

__global__ void k_dinv(float* __restrict__ deg) {
  int i = blockIdx.x * blockDim.x + threadIdx.x;
  if (i < N_NODES) deg[i] = rsqrtf(deg[i]);   // deg >= 1 always
}

__global__ void k_norm(const long long* __restrict__ src, const long long* __restrict__ dst,
                       const float* __restrict__ dinv, float* __restrict__ norm) {
  int e = blockIdx.x * blockDim.x + threadIdx.x;
  if (e < N_EDGES) norm[e] = dinv[(int)src[e]] * dinv[(int)dst[e]];
}

// ---------------- dense GEMM  H = X @ W  via V_WMMA_F32_16X16X4_F32 --------------------
// Block = 256 threads = 8 waves. Each wave computes one 16-row slab x all 96 cols.
// W staged transposed+padded in LDS: Wt[col*100 + k], so each lane's B fragment
// (K=k,k+1 at fixed col) is one aligned 8B ds_load_b64, spread across all 64 banks.
__global__ __launch_bounds__(256) void k_gemm(const float* __restrict__ X,
                                              const float* __restrict__ W,
                                              float* __restrict__ H) {
  __shared__ float Wt[HID * WT_STRIDE];
  for (int i = threadIdx.x; i < HID * HID; i += 256) {
    int k = i / HID, col = i - k * HID;
    Wt[col * WT_STRIDE + k] = W[i];
  }
  __syncthreads();

  const int lane = threadIdx.x & 31;
  const int wave = threadIdx.x >> 5;
  const int tile = blockIdx.x * 8 + wave;          // 16-row tile index
  if (tile >= N_NODES / 16) return;                // uniform per wave; EXEC stays all-1s
  const int row0 = tile * 16;

  const int m  = lane & 15;                        // A: M = lane&15 ; B/D: N = lane&15
  const int ko = (lane >> 4) << 1;                 // A/B: K sub-offset 0 or 2
  const float* xr = X + (size_t)(row0 + m) * HID;

  v8f acc[6] = {};                                 // 6 col-tiles of 16

  #pragma unroll 4
  for (int kb = 0; kb < HID; kb += 4) {
    const int k = kb + ko;
    // A fragment: row m, K = k..k+1 (8B-aligned: row stride 384B, k even)
    v2f a = *(const v2f*)(xr + k);
    #pragma unroll
    for (int t = 0; t < 6; ++t) {
      // B fragment: rows K=k,k+1 at col t*16+m -> contiguous in transposed LDS
      v2f b = *(const v2f*)(&Wt[(t * 16 + m) * WT_STRIDE + k]);
      acc[t] = __builtin_amdgcn_wmma_f32_16x16x4_f32(
          /*neg_a=*/false, a, /*neg_b=*/false, b,
          /*c_mod=*/(short)0, acc[t], /*reuse_a=*/false, /*reuse_b=*/false);
    }
  }

  // D layout: VGPR r -> row (r + 8*(lane>>4)), col = lane&15
  const int nlo = lane & 15;
  const int mh  = (lane >> 4) << 3;
  #pragma unroll
  for (int t = 0; t < 6; ++t)
    #pragma unroll
    for (int r = 0; r < 8; ++r)
      H[(size_t)(row0 + mh + r) * HID + t * 16 + nlo] = acc[t][r];
}

// ---------------- aggregation: out = selfloop + sum_E norm*h[src] ----------------------
__global__ void k_selfloop(const float* __restrict__ h, const float* __restrict__ dinv,
                           float* __restrict__ out) {
  int t = blockIdx.x * blockDim.x + threadIdx.x;
  if (t < N_NODES * HID) {
    float d = dinv[t / HID];
    out[t] = h[t] * d * d;
  }
}

// 24 threads per edge; each thread moves a float4 feature chunk.
__global__ void k_scatter(const float* __restrict__ h,
                          const long long* __restrict__ src,
                          const long long* __restrict__ dst,
                          const float* __restrict__ norm,
                          float* __restrict__ out) {
  int t = blockIdx.x * blockDim.x + threadIdx.x;
  if (t >= N_EDGES * 24) return;
  int e = t / 24;
  int c = (t % 24) << 2;                           // feature offset 0..92
  // stream the linear index/norm arrays ahead (global_prefetch_b8, speculative)
  if (e + 1024 < N_EDGES) {
    __builtin_prefetch(src + e + 1024, 0, 1);
    __builtin_prefetch(dst + e + 1024, 0, 1);
    __builtin_prefetch(norm + e + 1024, 0, 1);
  }
  int s = (int)src[e], d = (int)dst[e];
  float w = norm[e];
  const float4 v = *(const float4*)(h + (size_t)s * HID + c);
  float* o = out + (size_t)d * HID + c;
  atomic_add_f32(o + 0, v.x * w);
  atomic_add_f32(o + 1, v.y * w);
  atomic_add_f32(o + 2, v.z * w);
  atomic_add_f32(o + 3, v.w * w);
}

__global__ void k_bias_relu(float* __restrict__ x, const float* __restrict__ b) {
  int t = blockIdx.x * blockDim.x + threadIdx.x;
  if (t < N_NODES * HID) x[t] = fmaxf(x[t] + b[t % HID], 0.0f);
}

// ---------------- global mean pool + heads ---------------------------------------------
__global__ void k_zero_g(float* __restrict__ g) { g[threadIdx.x] = 0.0f; }

__global__ void k_pool(const float* __restrict__ h, float* __restrict__ g) {
  int f = threadIdx.x;                             // blockDim = HID
  float s = 0.0f;
  for (int i = blockIdx.x; i < N_NODES; i += gridDim.x) s += h[(size_t)i * HID + f];
  atomic_add_f32(&g[f], s);
}

__global__ void k_heads(const float* __restrict__ gsum,
                        const float* __restrict__ aW, const float* __restrict__ ab,
                        const float* __restrict__ cW, const float* __restrict__ cb,
                        float* __restrict__ out) {
  int j = blockIdx.x * blockDim.x + threadIdx.x;
  const float inv = 1.0f / (float)N_NODES;
  if (j < N_ACTIONS) {
    float s = ab[j];
    for (int f = 0; f < HID; ++f) s = fmaf(gsum[f] * inv, aW[f * N_ACTIONS + j], s);
    out[j] = s;
  } else if (j == N_ACTIONS) {
    float s = cb[0];
    for (int f = 0; f < HID; ++f) s = fmaf(gsum[f] * inv, cW[f], s);
    out[N_ACTIONS] = s;
  }
}

// ---------------------------------------------------------------------------------------
extern "C" void kernel_launch(void* const* d_in, const int* in_sizes, int n_in,
                              void* d_out, int out_size, void* d_ws, size_t ws_size,
                              hipStream_t stream) {
  (void)in_sizes; (void)n_in; (void)out_size; (void)ws_size;

  const float*     x      = (const float*)d_in[0];
  const long long* ei     = (const long long*)d_in[1];   // int64 [2, E]
  const float*     convW  = (const float*)d_in[2];       // [L, H, H]
  const float*     convB  = (const float*)d_in[3];       // [L, H]
  const float*     actorW = (const float*)d_in[4];       // [H, 729]
  const float*     actorB = (const float*)d_in[5];
  const float*     critW  = (const float*)d_in[6];       // [H, 1]
  const float*     critB  = (const float*)d_in[7];
  float*           out    = (float*)d_out;               // logits[729] ++ value

  const long long* src = ei;
  const long long* dst = ei + N_EDGES;

  // workspace carve-up (all fp32): dinv | norm | gsum(pad 128) | H | actA | actB
  float* dinv = (float*)d_ws;
  float* norm = dinv + N_NODES;
  float* gsum = norm + N_EDGES;
  float* bufH = gsum + 128;
  float* bufA = bufH + (size_t)N_NODES * HID;
  float* bufB = bufA + (size_t)N_NODES * HID;

  const int B = 256;
  // degree / per-edge norm (layer-invariant)
  k_deg_init <<<(N_NODES + B - 1) / B, B, 0, stream>>>(dinv);
  k_deg_accum<<<(N_EDGES + B - 1) / B, B, 0, stream>>>(dst, dinv);
  k_dinv     <<<(N_NODES + B - 1) / B, B, 0, stream>>>(dinv);
  k_norm     <<<(N_EDGES + B - 1) / B, B, 0, stream>>>(src, dst, dinv, norm);

  const float* cur = x;
  float* acts[2] = {bufA, bufB};
  const int gemm_blocks = (N_NODES / 16 + 7) / 8;        // 8 waves x 16 rows per block
  for (int l = 0; l < N_LAYERS; ++l) {
    float* o = acts[l & 1];
    k_gemm     <<<gemm_blocks, 256, 0, stream>>>(cur, convW + (size_t)l * HID * HID, bufH);
    k_selfloop <<<(N_NODES * HID + B - 1) / B, B, 0, stream>>>(bufH, dinv, o);
    k_scatter  <<<(N_EDGES * 24 + B - 1) / B, B, 0, stream>>>(bufH, src, dst, norm, o);
    k_bias_relu<<<(N_NODES * HID + B - 1) / B, B, 0, stream>>>(o, convB + (size_t)l * HID);
    cur = o;
  }

  k_zero_g<<<1, 128, 0, stream>>>(gsum);
  k_pool  <<<512, HID, 0, stream>>>(cur, gsum);
  k_heads <<<(N_ACTIONS + 1 + B - 1) / B, B, 0, stream>>>(gsum, actorW, actorB, critW, critB, out);
}